// RandomProjectionQuantizer_55070070669367
// MI455X (gfx1250) — compile-verified
//
#include <hip/hip_runtime.h>

typedef float v2f __attribute__((ext_vector_type(2)));
typedef float v8f __attribute__((ext_vector_type(8)));
typedef int   v4i __attribute__((ext_vector_type(4)));

#define AS1 __attribute__((address_space(1)))
#define AS3 __attribute__((address_space(3)))

#define INPUT_DIM 1024
#define CB_DIM 16
#define CB_SIZE 8192
#define NPOINTS 32768          // B*N = 16*2048
#define WAVES 8
#define PTS_PER_BLOCK 128      // WAVES * 16
#define KC_PROJ (INPUT_DIM / 4)    // 256 K-chunks of 4 for fp32 WMMA
#define CB_SLICE (CB_SIZE / WAVES) // 1024 codebook rows per wave
#define CB_TILES (CB_SLICE / 16)   // 64 tiles of 16 rows

__global__ __launch_bounds__(256) void rpq_wmma_kernel(
    const float* __restrict__ x,    // [32768, 1024]
    const float* __restrict__ rp,   // [1024, 16]
    const float* __restrict__ cb,   // [8192, 16]
    int* __restrict__ out)          // [32768]
{
  extern __shared__ float smem[];
  float* rp_s   = smem;                                   // 16384 f
  float* proj_s = rp_s + INPUT_DIM * CB_DIM;              // 2048 f
  float* red_v  = proj_s + PTS_PER_BLOCK * CB_DIM;        // 1024 f
  int*   red_i  = (int*)(red_v + WAVES * PTS_PER_BLOCK);  // 1024 i

  const int tid  = threadIdx.x;
  const int w    = tid >> 5;     // wave id 0..7
  const int lane = tid & 31;
  const int cL   = lane & 15;    // column / row-within-tile index
  const int s    = lane >> 4;    // half-wave selector (K/M high half)

  // ---- Stage random projection matrix into LDS ----
  // CDNA5 async DMA path: GLOBAL_LOAD_ASYNC_TO_LDS_B128 (ASYNCcnt-tracked),
  // then s_wait_asynccnt 0 before the block barrier.
#if __has_builtin(__builtin_amdgcn_global_load_async_to_lds_b128)
  {
    #pragma unroll
    for (int j = 0; j < (INPUT_DIM * CB_DIM / 4) / 256; ++j) {
      const float* g = rp + (size_t)(tid + j * 256) * 4;
      float*       l = rp_s + (size_t)(tid + j * 256) * 4;
      __builtin_amdgcn_global_load_async_to_lds_b128(
          (AS1 v4i*)g, (AS3 v4i*)l, 0, 0);
    }
  }
  #if __has_builtin(__builtin_amdgcn_s_wait_asynccnt)
    __builtin_amdgcn_s_wait_asynccnt(0);
  #else
    asm volatile("s_wait_asynccnt 0" ::: "memory");
  #endif
#else
  {
    const float4* g = (const float4*)rp;
    float4* l = (float4*)rp_s;
    #pragma unroll
    for (int j = 0; j < (INPUT_DIM * CB_DIM / 4) / 256; ++j)
      l[tid + j * 256] = g[tid + j * 256];
  }
#endif
  __syncthreads();

  // ---- Phase 1: projection tile (16 pts x 16 dims) via V_WMMA_F32_16X16X4_F32
  // A-frag (16x4 f32): lane<16 -> row cL, K = 4kc+{0,1}; lane>=16 -> row cL, K = 4kc+{2,3}
  // B-frag (4x16 f32): V0 = RP row (4kc+2s), col cL; V1 = next row
  const int ptbase = blockIdx.x * PTS_PER_BLOCK + w * 16;
  v8f pc = {0.f, 0.f, 0.f, 0.f, 0.f, 0.f, 0.f, 0.f};
  {
    const float* xrow = x + (size_t)(ptbase + cL) * INPUT_DIM + 2 * s;
    #pragma unroll 4
    for (int kc = 0; kc < KC_PROJ; ++kc) {
      __builtin_prefetch(xrow + 4 * kc + 256, 0, 3);   // stream-ahead prefetch
      v2f a = *(const v2f*)(xrow + 4 * kc);
      int e0 = 4 * kc + 2 * s;
      v2f b;
      b.x = rp_s[e0 * 16 + cL];
      b.y = rp_s[(e0 + 1) * 16 + cL];
      pc = __builtin_amdgcn_wmma_f32_16x16x4_f32(false, a, false, b,
                                                 (short)0, pc, false, false);
    }
  }
  // D layout: VGPR j, lane L -> (pt = j + 8*s, e = cL). Store proj[pt][e].
  #pragma unroll
  for (int j = 0; j < 8; ++j)
    proj_s[w * 256 + (j + 8 * s) * 16 + cL] = pc[j];
  __syncthreads();

  // ---- Phase 2: codebook sweep. B-frags = proj^T for all 8 point tiles.
  // B element (K=e, N=pt) = proj[pt][e] -> contiguous v2f at proj_s[pt*16 + e]
  v2f bfr[8][4];
  #pragma unroll
  for (int tau = 0; tau < 8; ++tau)
    #pragma unroll
    for (int ec = 0; ec < 4; ++ec)
      bfr[tau][ec] = *(const v2f*)(proj_s + tau * 256 + cL * 16 + 4 * ec + 2 * s);

  float bestv[8];
  int   besti[8];
  #pragma unroll
  for (int tau = 0; tau < 8; ++tau) { bestv[tau] = -__builtin_inff(); besti[tau] = 0; }

  const int kslice = w * CB_SLICE;
  for (int t = 0; t < CB_TILES; ++t) {
    const int k0 = kslice + t * 16;
    // A-frag: lane holds cb row (k0 + cL), cols 4kc + 2s + {0,1}
    const float* crow = cb + (size_t)(k0 + cL) * 16 + 2 * s;
    v2f a[4];
    #pragma unroll
    for (int kc = 0; kc < 4; ++kc) a[kc] = *(const v2f*)(crow + 4 * kc);
    // Row norm: this lane has 8 of 16 columns; partner lane (L^16) has the rest.
    float ss = 0.f;
    #pragma unroll
    for (int kc = 0; kc < 4; ++kc) ss += a[kc].x * a[kc].x + a[kc].y * a[kc].y;
    ss += __shfl_xor(ss, 16, 32);
    float rn = 1.0f / fmaxf(__builtin_sqrtf(ss), 1e-12f);
    #pragma unroll
    for (int kc = 0; kc < 4; ++kc) { a[kc].x *= rn; a[kc].y *= rn; }

    #pragma unroll
    for (int tau = 0; tau < 8; ++tau) {
      v8f acc = {0.f, 0.f, 0.f, 0.f, 0.f, 0.f, 0.f, 0.f};
      #pragma unroll
      for (int ec = 0; ec < 4; ++ec)
        acc = __builtin_amdgcn_wmma_f32_16x16x4_f32(false, a[ec], false,
                                                    bfr[tau][ec], (short)0, acc,
                                                    false, false);
      // D: VGPR j, lane L -> cb row k0 + j + 8*s, point = cL of tile tau
      #pragma unroll
      for (int j = 0; j < 8; ++j) {
        float v = acc[j];
        int k = k0 + j + 8 * s;
        bool better = v > bestv[tau];   // strict > keeps lowest index on ties
        bestv[tau] = better ? v : bestv[tau];
        besti[tau] = better ? k : besti[tau];
      }
    }
  }

  // Combine lane pair (L, L^16): each covers complementary cb-row halves
  #pragma unroll
  for (int tau = 0; tau < 8; ++tau) {
    float pv = __shfl_xor(bestv[tau], 16, 32);
    int   pi = __shfl_xor(besti[tau], 16, 32);
    bool take = (pv > bestv[tau]) || (pv == bestv[tau] && pi < besti[tau]);
    float fv = take ? pv : bestv[tau];
    int   fi = take ? pi : besti[tau];
    if (lane < 16) {
      red_v[w * 128 + tau * 16 + lane] = fv;
      red_i[w * 128 + tau * 16 + lane] = fi;
    }
  }
  __syncthreads();

  // Reduce across the 8 waves (each covered a distinct 1024-row cb slice)
  if (tid < PTS_PER_BLOCK) {
    float bv = red_v[tid];
    int   bi = red_i[tid];
    #pragma unroll
    for (int ww = 1; ww < WAVES; ++ww) {
      float v = red_v[ww * 128 + tid];
      int   i = red_i[ww * 128 + tid];
      if (v > bv || (v == bv && i < bi)) { bv = v; bi = i; }
    }
    out[blockIdx.x * PTS_PER_BLOCK + tid] = bi;
  }
}

extern "C" void kernel_launch(void* const* d_in, const int* in_sizes, int n_in,
                              void* d_out, int out_size, void* d_ws, size_t ws_size,
                              hipStream_t stream) {
  const float* x  = (const float*)d_in[0];   // [16, 2048, 1024]
  const float* rp = (const float*)d_in[1];   // [1024, 16]
  const float* cb = (const float*)d_in[2];   // [8192, 16]
  int* out = (int*)d_out;                    // [16, 2048] int32
  (void)in_sizes; (void)n_in; (void)out_size; (void)d_ws; (void)ws_size;

  const int blocks = NPOINTS / PTS_PER_BLOCK;          // 256
  const size_t shmem = (INPUT_DIM * CB_DIM             // rp stage
                        + PTS_PER_BLOCK * CB_DIM       // proj tiles
                        + WAVES * PTS_PER_BLOCK) * 4   // red_v
                       + WAVES * PTS_PER_BLOCK * 4;    // red_i  => 81920 B
  rpq_wmma_kernel<<<blocks, 256, shmem, stream>>>(x, rp, cb, out);
}